// AlgorithmAMultinomial_61933428415671
// MI455X (gfx1250) — compile-verified
//
#include <hip/hip_runtime.h>
#include <hip/hip_bf16.h>

// Exponential-race weighted sampling: per row, top-32 indices of log(u)/p.
// Memory-bound streaming kernel: 1.02 GB read once -> ~44us floor @ 23.3 TB/s.
// One workgroup (256 threads, 8 wave32) per row; LDS candidate filter keeps
// the per-element cost at one compare; survivors (~0.2%) go through
// ds_add_rtn_u32 append. Non-temporal b128 loads avoid thrashing the 192MB L2
// with a 1GB one-shot stream. 4 outstanding b128 loads per thread per chunk
// (~64KB in flight per WGP) to cover HBM latency at the per-WGP BW share.
// WMMA is inapplicable (selection, not GEMM).

#define TOPK 32
#define THREADS 256
#define QUAD 4                         // elements per coalesced b128 group
#define HALF (THREADS * QUAD)          // 1024: one coalesced region
#define CHUNK (2 * HALF)               // 2048 elements per chunk
#define TRIG 1024                      // rebuild when buffer >= TRIG
#define BUFCAP 3072                    // TRIG-1 + CHUNK = 3071 fits
#define POOLCAP (BUFCAP + TOPK)        // + current top-32 merged at rebuild

typedef float v4f __attribute__((ext_vector_type(4)));

// Monotone float -> uint mapping: a > b (float) <=> map(a) > map(b) (uint).
__device__ __forceinline__ unsigned sortable(float f) {
    unsigned b = __float_as_uint(f);
    return b ^ (unsigned)(((int)b >> 31) | 0x80000000);
}

// Merge candidate buffer with current top-32, extract new top-32 in
// descending order via iterated parallel argmax, tighten threshold, reset.
// Rare (O(1) calls/row), so simplicity > speed here.
__device__ __forceinline__ void rebuild_top(
    unsigned* s_keys, int* s_idx,
    unsigned* s_topk, int* s_topi,
    unsigned* s_redk, int* s_redp,
    unsigned* s_count, unsigned* s_thresh,
    bool have_top, int tid)
{
    __syncthreads();                       // all appends visible
    unsigned n = *s_count;
    if (n > BUFCAP) n = BUFCAP;
    if (have_top && tid < TOPK) {          // fold in current top-32
        s_keys[n + tid] = s_topk[tid];
        s_idx [n + tid] = s_topi[tid];
    }
    if (have_top) n += TOPK;
    __syncthreads();

    for (int s = 0; s < TOPK; ++s) {
        unsigned bk = 0u; int bp = -1;
        for (unsigned j = (unsigned)tid; j < n; j += THREADS) {
            unsigned k = s_keys[j];
            if (k > bk) { bk = k; bp = (int)j; }
        }
        s_redk[tid] = bk; s_redp[tid] = bp;
        __syncthreads();
        for (int off = THREADS / 2; off > 0; off >>= 1) {
            if (tid < off && s_redk[tid + off] > s_redk[tid]) {
                s_redk[tid] = s_redk[tid + off];
                s_redp[tid] = s_redp[tid + off];
            }
            __syncthreads();
        }
        if (tid == 0) {
            int p = s_redp[0];
            s_topk[s] = s_redk[0];
            s_topi[s] = (p >= 0) ? s_idx[p] : 0;
            if (p >= 0) s_keys[p] = 0u;    // extracted: remove from pool
        }
        __syncthreads();
    }
    if (tid == 0) { *s_thresh = s_topk[TOPK - 1]; *s_count = 0u; }
    __syncthreads();
}

__global__ __launch_bounds__(THREADS)
void AlgorithmAMultinomial_topk_kernel(const float* __restrict__ probs,
                                       const float* __restrict__ urnd,
                                       int* __restrict__ out, int V)
{
    __shared__ unsigned s_keys[POOLCAP];
    __shared__ int      s_idx [POOLCAP];
    __shared__ unsigned s_topk[TOPK];
    __shared__ int      s_topi[TOPK];
    __shared__ unsigned s_redk[THREADS];
    __shared__ int      s_redp[THREADS];
    __shared__ unsigned s_count;
    __shared__ unsigned s_thresh;

    const int tid = threadIdx.x;
    const int row = blockIdx.x;
    const float* __restrict__ prow = probs + (size_t)row * (size_t)V;
    const float* __restrict__ urow = urnd  + (size_t)row * (size_t)V;

    if (tid == 0) { s_count = 0u; s_thresh = 0u; }
    bool have_top = false;
    __syncthreads();

    const int vfull = (V / CHUNK) * CHUNK;   // V=128000 -> 62 chunks + 1024 tail
    for (int base = 0; base < vfull; base += CHUNK) {
        if (s_count >= TRIG) {               // uniform branch (post-barrier)
            rebuild_top(s_keys, s_idx, s_topk, s_topi, s_redk, s_redp,
                        &s_count, &s_thresh, have_top, tid);
            have_top = true;
        }
        const unsigned th = s_thresh;
        // Two coalesced regions per chunk: each b128 across a wave covers a
        // contiguous 512B span; 4 loads in flight per thread before any use.
        const int ia = base + tid * QUAD;
        const int ib = ia + HALF;
        v4f pa = __builtin_nontemporal_load((const v4f*)(prow + ia));
        v4f pb = __builtin_nontemporal_load((const v4f*)(prow + ib));
        v4f ua = __builtin_nontemporal_load((const v4f*)(urow + ia));
        v4f ub = __builtin_nontemporal_load((const v4f*)(urow + ib));
        #pragma unroll
        for (int c = 0; c < QUAD; ++c) {
            // key = log(u)/p via fast log + v_rcp_f32 (p in [1e-6,1), safe)
            float ka = __logf(ua[c]) * __builtin_amdgcn_rcpf(pa[c]);
            unsigned ma = sortable(ka);
            if (ma > th) {                   // ~0.2% of elements survive
                unsigned pos = atomicAdd(&s_count, 1u);
                if (pos < BUFCAP) { s_keys[pos] = ma; s_idx[pos] = ia + c; }
            }
            float kb = __logf(ub[c]) * __builtin_amdgcn_rcpf(pb[c]);
            unsigned mb = sortable(kb);
            if (mb > th) {
                unsigned pos = atomicAdd(&s_count, 1u);
                if (pos < BUFCAP) { s_keys[pos] = mb; s_idx[pos] = ib + c; }
            }
        }
        __syncthreads();                     // appends visible for next check
    }

    // Scalar tail (last 1024 elements for V=128000; ~0.8% of the traffic).
    if (vfull < V) {
        if (s_count >= TRIG) {
            rebuild_top(s_keys, s_idx, s_topk, s_topi, s_redk, s_redp,
                        &s_count, &s_thresh, have_top, tid);
            have_top = true;
        }
        const unsigned th = s_thresh;
        for (int i = vfull + tid; i < V; i += THREADS) {
            float key = __logf(urow[i]) * __builtin_amdgcn_rcpf(prow[i]);
            unsigned m = sortable(key);
            if (m > th) {
                unsigned pos = atomicAdd(&s_count, 1u);
                if (pos < BUFCAP) { s_keys[pos] = m; s_idx[pos] = i; }
            }
        }
    }

    // Final merge + extraction (descending order, matching lax.top_k).
    rebuild_top(s_keys, s_idx, s_topk, s_topi, s_redk, s_redp,
                &s_count, &s_thresh, have_top, tid);
    if (tid < TOPK) out[row * TOPK + tid] = s_topi[tid];
}

extern "C" void kernel_launch(void* const* d_in, const int* in_sizes, int n_in,
                              void* d_out, int out_size, void* d_ws, size_t ws_size,
                              hipStream_t stream) {
    const float* probs = (const float*)d_in[0];
    const float* urnd  = (const float*)d_in[1];
    int* out = (int*)d_out;                 // indices (int32), [B, 32] flat
    const int B = out_size / TOPK;          // 1024
    const int V = in_sizes[0] / B;          // 128000
    AlgorithmAMultinomial_topk_kernel<<<dim3(B), dim3(THREADS), 0, stream>>>(
        probs, urnd, out, V);
}